// TransformerCrossAttnLayer_38654705664445
// MI455X (gfx1250) — compile-verified
//
#include <hip/hip_runtime.h>
#include <stdint.h>

// Problem constants (from reference): W=256, HN=128, C=512, NHEAD=8, HD=64
#define W_  256
#define N_  128
#define C_  512
#define E_  8
#define HD_ 64

typedef float v2f __attribute__((ext_vector_type(2)));
typedef float v8f __attribute__((ext_vector_type(8)));

// ---------------- CDNA5 async global->LDS copy (ASYNCcnt path) ----------------
// LDS destination offset is passed in a VGPR; generic LDS pointer carries the LDS offset in addr[31:0]
__device__ inline void async_ld_b32(void* lds, const float* g) {
  asm volatile("global_load_async_to_lds_b32 %0, %1, off"
               :: "v"((uint32_t)(uintptr_t)lds), "v"(g)
               : "memory");
}
__device__ inline void wait_async0() {
  asm volatile("s_wait_asynccnt 0" ::: "memory");
}

// ---------------- wave32 reductions ----------------
__device__ inline float wave_sum(float x) {
#pragma unroll
  for (int o = 16; o > 0; o >>= 1) x += __shfl_xor(x, o, 32);
  return x;
}
__device__ inline float wave_max(float x) {
#pragma unroll
  for (int o = 16; o > 0; o >>= 1) x = fmaxf(x, __shfl_xor(x, o, 32));
  return x;
}

// ---------------- LayerNorm over C=512, one row per block ----------------
__global__ __launch_bounds__(256) void ln_kernel(
    const float* __restrict__ left, const float* __restrict__ right,
    const float* __restrict__ lw, const float* __restrict__ lb,
    float* __restrict__ outL, float* __restrict__ outR) {
  const int row  = blockIdx.x;
  const float* x = (blockIdx.y ? right : left) + (size_t)row * C_;
  float* o       = (blockIdx.y ? outR : outL) + (size_t)row * C_;
  const int t = threadIdx.x, lane = t & 31, wave = t >> 5;
  __shared__ float red[8];

  float x0 = x[t], x1 = x[t + 256];
  float s = wave_sum(x0 + x1);
  if (!lane) red[wave] = s;
  __syncthreads();
  float mean = 0.f;
#pragma unroll
  for (int i = 0; i < 8; ++i) mean += red[i];
  mean *= (1.0f / C_);
  __syncthreads();

  float d0 = x0 - mean, d1 = x1 - mean;
  float vs = wave_sum(d0 * d0 + d1 * d1);
  if (!lane) red[wave] = vs;
  __syncthreads();
  float var = 0.f;
#pragma unroll
  for (int i = 0; i < 8; ++i) var += red[i];
  var *= (1.0f / C_);
  float rstd = rsqrtf(var + 1e-5f);

  o[t]       = d0 * rstd * lw[t]       + lb[t];
  o[t + 256] = d1 * rstd * lw[t + 256] + lb[t + 256];
}

// ---------------- Generic strided/batched fp32 WMMA GEMM ----------------
// C[m,n](+)= scale * (sum_k A[m,k]*B[n,k] + bias[n]) + residual
// A element:  A[a_off + rowSel(m)*a_row + k*a_k]   (rowSel via optional gather)
// B element:  B[b_off + n*b_row + k*b_k]
// C element:  C[c_off + m*c_row + n*c_col]
// batch z -> (i0 = z/d1, i1 = z%d1); *_off = i0*(_b0) + i1*(_b1)
// Block tile 128x64, BK=32, 8 wave32s each owning a 32x32 sub-tile (4 WMMA accs).
// Global->LDS staging via async-to-LDS with double buffering.
// Requirements: M % 128 == 0, Nc % 64 == 0, K % 32 == 0 (all satisfied here).
__global__ __launch_bounds__(256) void gemm_f32_wmma(
    const float* __restrict__ A, const float* __restrict__ B, float* __restrict__ Cm,
    const int* __restrict__ gather, const float* __restrict__ bias,
    const float* __restrict__ residual,
    int M, int Nc, int K,
    long long a_row, long long a_k, long long b_row, long long b_k,
    long long c_row, long long c_col,
    long long a_b0, long long a_b1, long long b_b0, long long b_b1,
    long long c_b0, long long c_b1,
    int batch_d1, float scale, int accumulate) {
  __shared__ float As[2][128][33];
  __shared__ float Bs[2][64][33];

  const int z  = blockIdx.z;
  const int i0 = z / batch_d1, i1 = z - i0 * batch_d1;
  const long long a_off = (long long)i0 * a_b0 + (long long)i1 * a_b1;
  const long long b_off = (long long)i0 * b_b0 + (long long)i1 * b_b1;
  const long long c_off = (long long)i0 * c_b0 + (long long)i1 * c_b1;

  const int tid  = threadIdx.x;
  const int lane = tid & 31, wave = tid >> 5;
  const int half = lane >> 4, l15 = lane & 15;
  const int m0 = (wave & 3) << 5;   // wave's 32-row strip within 128-row tile
  const int n0 = (wave >> 2) << 5;  // wave's 32-col strip within 64-col tile
  const int blockM = blockIdx.y << 7;
  const int blockN = blockIdx.x << 6;

  v8f acc00 = {0.f, 0.f, 0.f, 0.f, 0.f, 0.f, 0.f, 0.f};
  v8f acc01 = acc00, acc10 = acc00, acc11 = acc00;

  // issue one K-tile's async copies into buffer `buf`
  auto issue_tile = [&](int k0, int buf) {
#pragma unroll
    for (int i = 0; i < 16; ++i) {          // A: 128x32 = 4096 elems / 256 thr
      int idx = tid + i * 256;
      int r = idx >> 5, c = idx & 31;
      int gm = blockM + r;
      long long rowSel = gather ? (long long)gather[gm] : (long long)gm;
      async_ld_b32(&As[buf][r][c],
                   &A[a_off + rowSel * a_row + (long long)(k0 + c) * a_k]);
    }
#pragma unroll
    for (int i = 0; i < 8; ++i) {           // B: 64x32 = 2048 elems / 256 thr
      int idx = tid + i * 256;
      int r = idx >> 5, c = idx & 31;
      async_ld_b32(&Bs[buf][r][c],
                   &B[b_off + (long long)(blockN + r) * b_row +
                      (long long)(k0 + c) * b_k]);
    }
  };

  int cur = 0;
  issue_tile(0, cur);
  wait_async0();
  __syncthreads();

  for (int k0 = 0; k0 < K; k0 += 32) {
    int nxt = cur ^ 1;
    if (k0 + 32 < K) issue_tile(k0 + 32, nxt);  // overlap next tile with compute

#pragma unroll
    for (int kk = 0; kk < 32; kk += 4) {
      // fp32 WMMA 16x16x4 fragments: lanes 0-15 carry K=kk,kk+1; lanes 16-31 K=kk+2,kk+3
      v2f a0, a1, b0, b1;
      a0.x = As[cur][m0 + l15][kk + 2 * half];
      a0.y = As[cur][m0 + l15][kk + 2 * half + 1];
      a1.x = As[cur][m0 + 16 + l15][kk + 2 * half];
      a1.y = As[cur][m0 + 16 + l15][kk + 2 * half + 1];
      b0.x = Bs[cur][n0 + l15][kk + 2 * half];
      b0.y = Bs[cur][n0 + l15][kk + 2 * half + 1];
      b1.x = Bs[cur][n0 + 16 + l15][kk + 2 * half];
      b1.y = Bs[cur][n0 + 16 + l15][kk + 2 * half + 1];
      acc00 = __builtin_amdgcn_wmma_f32_16x16x4_f32(false, a0, false, b0, (short)0, acc00, false, false);
      acc01 = __builtin_amdgcn_wmma_f32_16x16x4_f32(false, a0, false, b1, (short)0, acc01, false, false);
      acc10 = __builtin_amdgcn_wmma_f32_16x16x4_f32(false, a1, false, b0, (short)0, acc10, false, false);
      acc11 = __builtin_amdgcn_wmma_f32_16x16x4_f32(false, a1, false, b1, (short)0, acc11, false, false);
    }

    wait_async0();
    __syncthreads();
    cur = nxt;
  }

  // D layout: VGPR r holds M = r (lanes 0-15) / r+8 (lanes 16-31), N = lane&15
  auto store_acc = [&](const v8f& acc, int mbase, int nbase) {
#pragma unroll
    for (int r = 0; r < 8; ++r) {
      int gm  = blockM + mbase + r + 8 * half;
      int col = blockN + nbase + l15;
      float v = acc[r];
      if (bias) v += bias[col];
      v *= scale;
      long long ci = c_off + (long long)gm * c_row + (long long)col * c_col;
      if (residual) v += residual[ci];
      if (accumulate) Cm[ci] += v;
      else            Cm[ci] = v;
    }
  };
  store_acc(acc00, m0,      n0);
  store_acc(acc01, m0,      n0 + 16);
  store_acc(acc10, m0 + 16, n0);
  store_acc(acc11, m0 + 16, n0 + 16);
}

// ---------------- softmax over last axis + raw_attn (sum over heads) ----------------
__global__ __launch_bounds__(256) void softmax_raw_kernel(float* __restrict__ attn,
                                                          float* __restrict__ raw) {
  const int w = blockIdx.x, n = blockIdx.y;
  const int t = threadIdx.x, lane = t & 31, wave = t >> 5;
  __shared__ float red[8];
  float racc = 0.f;
  for (int e = 0; e < E_; ++e) {
    long long idx = ((((long long)n * E_ + e) * W_ + w) * W_) + t;
    float x = attn[idx];
    racc += x;
    float m = wave_max(x);
    if (!lane) red[wave] = m;
    __syncthreads();
    m = red[0];
#pragma unroll
    for (int i = 1; i < 8; ++i) m = fmaxf(m, red[i]);
    __syncthreads();
    float ex = expf(x - m);
    float s = wave_sum(ex);
    if (!lane) red[wave] = s;
    __syncthreads();
    float tot = 0.f;
#pragma unroll
    for (int i = 0; i < 8; ++i) tot += red[i];
    __syncthreads();
    attn[idx] = ex / tot;
  }
  raw[((long long)n * W_ + w) * W_ + t] = racc;
}

// ---------------- host-side launcher ----------------
static inline void launch_gemm(hipStream_t s, const float* A, const float* B, float* Cc,
                               const int* gather, const float* bias, const float* residual,
                               int M, int Nc, int K,
                               long long a_row, long long a_k, long long b_row, long long b_k,
                               long long c_row, long long c_col,
                               long long a_b0, long long a_b1, long long b_b0, long long b_b1,
                               long long c_b0, long long c_b1,
                               int nbatch, int d1, float scale, int accumulate) {
  dim3 grid(Nc / 64, M / 128, nbatch);
  gemm_f32_wmma<<<grid, 256, 0, s>>>(A, B, Cc, gather, bias, residual, M, Nc, K,
                                     a_row, a_k, b_row, b_k, c_row, c_col,
                                     a_b0, a_b1, b_b0, b_b1, c_b0, c_b1,
                                     d1, scale, accumulate);
}

extern "C" void kernel_launch(void* const* d_in, const int* in_sizes, int n_in,
                              void* d_out, int out_size, void* d_ws, size_t ws_size,
                              hipStream_t stream) {
  const float* feat_left  = (const float*)d_in[0];
  const float* feat_right = (const float*)d_in[1];
  const float* pos        = (const float*)d_in[2];
  const int*   pos_idx    = (const int*)d_in[3];
  const float* ln_w       = (const float*)d_in[4];
  const float* ln_b       = (const float*)d_in[5];
  const float* in_w       = (const float*)d_in[6];  // [3C, C]
  const float* in_b       = (const float*)d_in[7];  // [3C]
  const float* out_w      = (const float*)d_in[8];  // [C, C]
  const float* out_b      = (const float*)d_in[9];  // [C]

  const size_t SZ_WNC = (size_t)W_ * N_ * C_;        // 16.78M floats
  const size_t SZ_WWC = (size_t)W_ * W_ * C_;        // 33.55M floats
  const size_t SZ_ATT = (size_t)N_ * E_ * W_ * W_;   // 67.11M floats
  // workspace layout (~941 MB total)
  float* ws   = (float*)d_ws;
  float* qln  = ws;
  float* kln  = qln + SZ_WNC;
  float* qb   = kln + SZ_WNC;
  float* kb   = qb  + SZ_WNC;
  float* vb   = kb  + SZ_WNC;
  float* qr   = vb  + SZ_WNC;
  float* kr   = qr  + SZ_WWC;
  float* attn = kr  + SZ_WWC;
  float* vo   = attn + SZ_ATT;

  const int MWN = W_ * N_;                  // 32768 rows
  const long long NC  = (long long)N_ * C_; // 65536
  const long long WW  = (long long)W_ * W_; // 65536
  const long long EWW = (long long)E_ * WW; // 524288
  const float scal = 0.125f;                // hd^-0.5

  // 1) LayerNorm of both streams
  ln_kernel<<<dim3(MWN, 2), 256, 0, stream>>>(feat_left, feat_right, ln_w, ln_b, qln, kln);

  // 2) q = (LN(left) @ Wq^T + bq) * scal
  launch_gemm(stream, qln, in_w, qb, nullptr, in_b, nullptr,
              MWN, C_, C_, C_, 1, C_, 1, C_, 1, 0, 0, 0, 0, 0, 0, 1, 1, scal, 0);
  // 3) k = LN(right) @ Wk^T + bk ; 4) v = LN(right) @ Wv^T + bv
  launch_gemm(stream, kln, in_w + (size_t)C_ * C_, kb, nullptr, in_b + C_, nullptr,
              MWN, C_, C_, C_, 1, C_, 1, C_, 1, 0, 0, 0, 0, 0, 0, 1, 1, 1.f, 0);
  launch_gemm(stream, kln, in_w + (size_t)2 * C_ * C_, vb, nullptr, in_b + 2 * C_, nullptr,
              MWN, C_, C_, C_, 1, C_, 1, C_, 1, 0, 0, 0, 0, 0, 0, 1, 1, 1.f, 0);

  // 5) q_r = (pos[idx] @ Wq^T + bq) * scal ; 6) k_r = pos[idx] @ Wk^T + bk
  launch_gemm(stream, pos, in_w, qr, pos_idx, in_b, nullptr,
              W_ * W_, C_, C_, C_, 1, C_, 1, C_, 1, 0, 0, 0, 0, 0, 0, 1, 1, scal, 0);
  launch_gemm(stream, pos, in_w + (size_t)C_ * C_, kr, pos_idx, in_b + C_, nullptr,
              W_ * W_, C_, C_, C_, 1, C_, 1, C_, 1, 0, 0, 0, 0, 0, 0, 1, 1, 1.f, 0);

  // 7) cc: attn[n,e,w,v] = Q_ne[w,:] . K_ne[v,:]   (batch over (n,e))
  launch_gemm(stream, qb, kb, attn, nullptr, nullptr, nullptr,
              W_, W_, HD_,
              NC, 1, NC, 1, W_, 1,
              C_, HD_, C_, HD_, EWW, WW,
              N_ * E_, E_, 1.f, 0);
  // 8) cp: attn[n,e,w,v] += Q[w][n,:] . k_r[w][v,:]  (batch over (w,e))
  launch_gemm(stream, qb, kr, attn, nullptr, nullptr, nullptr,
              N_, W_, HD_,
              C_, 1, C_, 1, EWW, 1,
              NC, HD_, (long long)W_ * C_, HD_, W_, WW,
              W_ * E_, E_, 1.f, 1);
  // 9) pc: attn[n,e,w,v] += K[v][n,:] . q_r[w,v,:]   (batch over (v,e); cols map to w)
  launch_gemm(stream, kb, qr, attn, nullptr, nullptr, nullptr,
              N_, W_, HD_,
              C_, 1, (long long)W_ * C_, 1, EWW, W_,
              NC, HD_, C_, HD_, 1, WW,
              W_ * E_, E_, 1.f, 1);

  // 10) raw_attn (sum over e) + in-place softmax over v
  softmax_raw_kernel<<<dim3(W_, N_), 256, 0, stream>>>(attn, (float*)d_out + SZ_WNC);

  // 11) pv: vo[w,n,e,:] = P_ne[w,:] @ V_ne  (batch over (n,e); B K-strided by NC)
  launch_gemm(stream, attn, vb, vo, nullptr, nullptr, nullptr,
              W_, HD_, W_,
              W_, 1, 1, NC, NC, 1,
              EWW, WW, C_, HD_, C_, HD_,
              N_ * E_, E_, 1.f, 0);

  // 12) out = vo @ out_w^T + out_b + feat_left  (fused residual)
  launch_gemm(stream, vo, out_w, (float*)d_out, nullptr, out_b, feat_left,
              MWN, C_, C_, C_, 1, C_, 1, C_, 1, 0, 0, 0, 0, 0, 0, 1, 1, 1.f, 0);
}